// MMD_loss_34626026340428
// MI455X (gfx1250) — compile-verified
//
#include <hip/hip_runtime.h>
#include <hip/hip_bf16.h>

#define NROWS 8192          // 2n
#define NHALF 4096          // n
#define KDIM  512
#define TILE  128
#define KC    32
#define LDS_STRIDE (KC + 4) // 36 floats: 32 data + 4 pad (matches TDM pad descriptor)

typedef float v2f __attribute__((ext_vector_type(2)));
typedef float v8f __attribute__((ext_vector_type(8)));
typedef unsigned int v4u __attribute__((ext_vector_type(4)));
typedef int v4i __attribute__((ext_vector_type(4)));
typedef int v8i __attribute__((ext_vector_type(8)));

__device__ __forceinline__ const float* row_ptr(const float* __restrict__ s,
                                                const float* __restrict__ t, int r) {
    return (r < NHALF) ? (s + (size_t)r * KDIM) : (t + (size_t)(r - NHALF) * KDIM);
}

// ---------------------------------------------------------------------------
// Tensor Data Mover: DMA a 128-row x 32-col f32 tile (row stride 512 floats)
// from global memory into LDS at byte offset lds_off, padding each 32-DWORD
// row with 4 DWORDs (-> 36-float LDS row stride). Descriptor per ISA 8.3/8.4.
// ---------------------------------------------------------------------------
__device__ __forceinline__ void tdm_tile_load(const float* gptr, unsigned lds_off) {
    const unsigned long long ga = (unsigned long long)(uintptr_t)gptr;
    v4u g0;
    g0.x = 1u;                                    // count=1 (valid user descriptor)
    g0.y = lds_off;                               // lds_addr (bytes)
    g0.z = (unsigned)(ga & 0xffffffffu);          // global_addr[31:0]
    g0.w = (unsigned)((ga >> 32) & 0x01ffffffu)   // global_addr[56:32]
         | (2u << 30);                            // type = 2 ("image")
    v8i g1;
    g1[0] = (2 << 16)      // data_size = 2 -> 4 bytes
          | (1 << 20)      // pad_enable
          | (4 << 22)      // pad_interval = 4 -> every 32 DWORDs
          | (3 << 25);     // pad_amount   = 3 -> 4 DWORDs
    g1[1] = (KDIM << 16);        // tensor_dim0 = 512 (bits 79:48, low half)
    g1[2] = (int)(4096u << 16);  // tensor_dim1 = 4096 (bits 111:80, low half)
    g1[3] = (KC << 16);          // tile_dim0 = 32  (bits 127:112)
    g1[4] = TILE;                // tile_dim1 = 128 (bits 143:128)
    g1[5] = KDIM;                // tensor_dim0_stride = 512 (bits 207:160, low)
    g1[6] = 0;
    g1[7] = 0;
    const v4i z4 = {0, 0, 0, 0};
#if __has_include(<hip/amd_detail/amd_gfx1250_TDM.h>)
    const v8i z8 = {0, 0, 0, 0, 0, 0, 0, 0};
    __builtin_amdgcn_tensor_load_to_lds(g0, g1, z4, z4, z8, 0);
#else
    __builtin_amdgcn_tensor_load_to_lds(g0, g1, z4, z4, 0);
#endif
}

// ---------------------------------------------------------------------------
// Kernel A: sq[i] = sum_k total[i][k]^2 ; also zero the two global accumulators
// ---------------------------------------------------------------------------
__global__ void mmd_rowsumsq_kernel(const float* __restrict__ src,
                                    const float* __restrict__ tgt,
                                    float* __restrict__ sq,
                                    float* __restrict__ accum) {
    if (blockIdx.x == 0 && threadIdx.x == 0) { accum[0] = 0.0f; accum[1] = 0.0f; }
    const int wave = threadIdx.x >> 5;   // 8 waves per block, one row each
    const int lane = threadIdx.x & 31;
    const int row  = blockIdx.x * 8 + wave;
    const float* p = row_ptr(src, tgt, row);
    float s = 0.0f;
    #pragma unroll
    for (int c = 0; c < KDIM / 32; ++c) {
        float v = p[lane + c * 32];
        s += v * v;
    }
    #pragma unroll
    for (int off = 16; off > 0; off >>= 1) s += __shfl_down(s, off, 32);
    if (lane == 0) sq[row] = s;
}

// ---------------------------------------------------------------------------
// Kernels B/D: f32-WMMA Gram tile with TDM double-buffered staging + epilogue.
//   PASS 1: accum[0] += sum( max(l2, 0) )            (for bandwidth)
//   PASS 2: accum[1] += sum( sign * Σ_i exp(-l2c/(bw*2^i)) )
// ---------------------------------------------------------------------------
template <int PASS>
__global__ __launch_bounds__(256)
void mmd_tile_kernel(const float* __restrict__ src, const float* __restrict__ tgt,
                     const float* __restrict__ sq,  const float* __restrict__ bw_ptr,
                     float* __restrict__ accum) {
    __shared__ float ldsA[2][TILE * LDS_STRIDE];
    __shared__ float ldsB[2][TILE * LDS_STRIDE];
    __shared__ float red[256];

    const int rowBase = blockIdx.y * TILE;
    const int colBase = blockIdx.x * TILE;
    const int tid    = threadIdx.x;
    const int wave   = tid >> 5;
    const int lane   = tid & 31;
    const int waveM  = wave >> 1;      // 0..3  -> 32-row strip
    const int waveN  = wave & 1;       // 0..1  -> 64-col strip
    const int lane16 = lane & 15;
    const int laneHi = lane >> 4;      // 0/1
    const int klane  = laneHi * 2;     // K offset within 4-wide slab

    const float* aBase = row_ptr(src, tgt, rowBase);  // 128-row tiles never straddle halves
    const float* bBase = row_ptr(src, tgt, colBase);

    v8f acc[2][4];
    #pragma unroll
    for (int mi = 0; mi < 2; ++mi)
        #pragma unroll
        for (int ni = 0; ni < 4; ++ni)
            acc[mi][ni] = 0.0f;

    // Prologue: DMA chunk 0 into buffer 0 (wave 0 drives the TDM).
    if (wave == 0) {
        tdm_tile_load(aBase, (unsigned)(uintptr_t)&ldsA[0][0]);
        tdm_tile_load(bBase, (unsigned)(uintptr_t)&ldsB[0][0]);
    }

    int buf = 0;
    for (int kc = 0; kc < KDIM; kc += KC, buf ^= 1) {
        if (wave == 0) {
            if (kc + KC < KDIM) {
                // Kick chunk kc+KC into the other buffer, then wait only for
                // chunk kc (2 older tensor ops may retire, 2 newer in flight).
                tdm_tile_load(aBase + (kc + KC), (unsigned)(uintptr_t)&ldsA[buf ^ 1][0]);
                tdm_tile_load(bBase + (kc + KC), (unsigned)(uintptr_t)&ldsB[buf ^ 1][0]);
                __builtin_amdgcn_s_wait_tensorcnt((short)2);
            } else {
                __builtin_amdgcn_s_wait_tensorcnt((short)0);
            }
        }
        __syncthreads();   // chunk kc visible to all 8 waves

        const float* lA = &ldsA[buf][0];
        const float* lB = &ldsB[buf][0];
        #pragma unroll
        for (int k8 = 0; k8 < KC / 4; ++k8) {
            const int kk = k8 * 4 + klane;
            v2f afrag[2], bfrag[4];
            #pragma unroll
            for (int mi = 0; mi < 2; ++mi) {
                const int M = waveM * 32 + mi * 16 + lane16;
                afrag[mi] = *(const v2f*)&lA[M * LDS_STRIDE + kk];
            }
            #pragma unroll
            for (int ni = 0; ni < 4; ++ni) {
                const int N = waveN * 64 + ni * 16 + lane16;
                bfrag[ni] = *(const v2f*)&lB[N * LDS_STRIDE + kk];
            }
            #pragma unroll
            for (int mi = 0; mi < 2; ++mi)
                #pragma unroll
                for (int ni = 0; ni < 4; ++ni)
                    acc[mi][ni] = __builtin_amdgcn_wmma_f32_16x16x4_f32(
                        false, afrag[mi], false, bfrag[ni],
                        (short)0, acc[mi][ni], false, false);
        }
        __syncthreads();   // done reading buf before TDM refills it (2 chunks later)
    }

    // ---- fused epilogue: l2 = sq[i] + sq[j] - 2*G ----
    float inv0 = 0.0f;
    if (PASS == 2) inv0 = 1.0f / bw_ptr[0];
    float local = 0.0f;

    #pragma unroll
    for (int mi = 0; mi < 2; ++mi) {
        const int tr = rowBase + waveM * 32 + mi * 16 + laneHi * 8; // C/D layout: VGPR r -> M = r + 8*laneHi
        float sqr[8];
        #pragma unroll
        for (int r = 0; r < 8; ++r) sqr[r] = sq[tr + r];
        const float rsgn = (tr < NHALF) ? 1.0f : -1.0f;
        #pragma unroll
        for (int ni = 0; ni < 4; ++ni) {
            const int tc = colBase + waveN * 64 + ni * 16 + lane16;
            const float sqc = sq[tc];
            const float sgn = rsgn * ((tc < NHALF) ? 1.0f : -1.0f);
            #pragma unroll
            for (int r = 0; r < 8; ++r) {
                const float l2 = sqr[r] + sqc - 2.0f * acc[mi][ni][r];
                if (PASS == 1) {
                    local += fmaxf(l2, 0.0f);
                } else {
                    const float l2c = fminf(fmaxf(l2, 0.0f), 1000.0f);
                    float kz = 0.0f, inv = inv0;
                    #pragma unroll
                    for (int i = 0; i < 5; ++i) { kz += __expf(-l2c * inv); inv *= 0.5f; }
                    local += sgn * kz;
                }
            }
        }
    }

    red[tid] = local;
    __syncthreads();
    #pragma unroll
    for (int s = 128; s > 0; s >>= 1) {
        if (tid < s) red[tid] += red[tid + s];
        __syncthreads();
    }
    if (tid == 0) atomicAdd(&accum[PASS - 1], red[0]);
}

// ---------------------------------------------------------------------------
// Kernel C: bandwidth = clip(sum_l2 / (N^2 - N), 0.001, 1000)
// ---------------------------------------------------------------------------
__global__ void mmd_bandwidth_kernel(const float* __restrict__ accum,
                                     float* __restrict__ bw) {
    const float denom = (float)NROWS * (float)NROWS - (float)NROWS; // exact in f32
    float b = accum[0] / denom;
    bw[0] = fminf(fmaxf(b, 0.001f), 1000.0f);
}

// ---------------------------------------------------------------------------
// Kernel E: out = signed_sum / n^2
// ---------------------------------------------------------------------------
__global__ void mmd_finalize_kernel(const float* __restrict__ accum,
                                    float* __restrict__ out) {
    out[0] = accum[1] * (1.0f / ((float)NHALF * (float)NHALF));
}

extern "C" void kernel_launch(void* const* d_in, const int* in_sizes, int n_in,
                              void* d_out, int out_size, void* d_ws, size_t ws_size,
                              hipStream_t stream) {
    const float* src = (const float*)d_in[0];
    const float* tgt = (const float*)d_in[1];
    float* out = (float*)d_out;

    // workspace layout: sq[8192] | accum[2] | bw[1]
    float* sq    = (float*)d_ws;
    float* accum = sq + NROWS;
    float* bw    = accum + 2;

    mmd_rowsumsq_kernel<<<NROWS / 8, 256, 0, stream>>>(src, tgt, sq, accum);

    dim3 grid(NROWS / TILE, NROWS / TILE);   // 64 x 64 tiles
    mmd_tile_kernel<1><<<grid, 256, 0, stream>>>(src, tgt, sq, bw, accum);
    mmd_bandwidth_kernel<<<1, 1, 0, stream>>>(accum, bw);
    mmd_tile_kernel<2><<<grid, 256, 0, stream>>>(src, tgt, sq, bw, accum);
    mmd_finalize_kernel<<<1, 1, 0, stream>>>(accum, out);
}